// GraphSAGELayer_67233418051656
// MI455X (gfx1250) — compile-verified
//
#include <hip/hip_runtime.h>

// ---------------------------------------------------------------------------
// GraphSAGE (mean) + LayerNorm + ReLU for MI455X (gfx1250, wave32).
//   N=50000 nodes, E=800000 edges, D_in = D_out = 64, all fp32.
// Roofline: edge scatter (~205MB L2 gathers + 51M f32 atomics) dominates;
// GEMM is 0.8 GFLOP / ~38MB -> fused FP32 WMMA (16x16x4) epilogue kernel.
// ---------------------------------------------------------------------------

typedef float v2f __attribute__((ext_vector_type(2)));
typedef float v8f __attribute__((ext_vector_type(8)));

#define LN_EPS 1e-5f

// ---------------- kernel 1: zero the scatter workspace ----------------------
__global__ void zero_ws_kernel(float* __restrict__ ws, long long n) {
    long long i = (long long)blockIdx.x * blockDim.x + threadIdx.x;
    if (i < n) ws[i] = 0.0f;
}

// ---------------- kernel 2: edge gather + atomic scatter --------------------
// One thread per (edge, feature): fully coalesced 4B gather from x[src] row
// (L2-resident, 12.8MB) and global_atomic_add_f32 into aggsum[dst] row.
__global__ void __launch_bounds__(256)
edge_scatter_kernel(const float* __restrict__ x,
                    const long long* __restrict__ ei,   // [2, E] int64
                    float* __restrict__ aggsum,         // [N, 64]
                    float* __restrict__ cnt,            // [N]
                    int E) {
    long long tid = (long long)blockIdx.x * blockDim.x + threadIdx.x;
    int e = (int)(tid >> 6);
    int f = (int)(tid & 63);
    if (e >= E) return;
    int src = (int)ei[e];
    int dst = (int)ei[(long long)E + e];
    float v = x[(long long)src * 64 + f];
    atomicAdd(&aggsum[(long long)dst * 64 + f], v);
    if (f == 0) atomicAdd(&cnt[dst], 1.0f);
}

// ---------------- kernel 3: fused (agg@Wl^T + x@Wr^T + b) -> LN -> ReLU -----
// One wave32 per 16-node tile. 4 n-tiles of 16x16 f32 accumulators.
// 16 K-steps of V_WMMA_F32_16X16X4_F32, both GEMMs into the same accumulators.
//
// 32-bit WMMA layouts (ISA 7.12.2):
//   A 16x4 : lanes 0-15 -> M=lane, VGPR{0,1} = K{0,1}; lanes 16-31 -> K{2,3}
//   B 4x16 : VGPR0 = K rows {0 | 2} over lane halves, VGPR1 = {1 | 3}; N=lane&15
//   C 16x16: VGPR r -> rows {r | r+8} over lane halves, N = lane&15
__global__ void __launch_bounds__(256)
sage_gemm_ln_kernel(const float* __restrict__ x,
                    const float* __restrict__ Wl,   // [64 out, 64 in]
                    const float* __restrict__ Wr,   // [64 out, 64 in]
                    const float* __restrict__ bias, // [64]
                    const float* __restrict__ lnw,  // [64]
                    const float* __restrict__ lnb,  // [64]
                    const float* __restrict__ aggsum, // [N, 64]
                    const float* __restrict__ cnt,    // [N]
                    float* __restrict__ out,          // [N, 64]
                    int N) {
    const int lane   = threadIdx.x & 31;
    const int waveIb = threadIdx.x >> 5;
    const int tile   = blockIdx.x * (blockDim.x >> 5) + waveIb;
    const int ntile  = (N + 15) >> 4;
    if (tile >= ntile) return;            // whole-wave uniform exit: EXEC stays full

    const int lo16 = lane & 15;           // 0..15
    const int hi   = lane >> 4;           // 0 or 1 (selects K pair / row half)

    int rowA = tile * 16 + lo16;
    if (rowA >= N) rowA = N - 1;          // N % 16 == 0 here; kept for safety

    const float invc = 1.0f / fmaxf(cnt[rowA], 1.0f);
    const float* __restrict__ aggRow = aggsum + (long long)rowA * 64;
    const float* __restrict__ xRow   = x      + (long long)rowA * 64;

    v8f acc[4] = {};   // 4 n-tiles of 16x16 f32, zero-initialized

    for (int kk = 0; kk < 16; ++kk) {
        const int c0 = kk * 4 + hi * 2;   // this lane's K pair within the K=4 step
        v2f aA, aX;
        aA[0] = aggRow[c0] * invc;  aA[1] = aggRow[c0 + 1] * invc;
        aX[0] = xRow[c0];           aX[1] = xRow[c0 + 1];
#pragma unroll
        for (int nt = 0; nt < 4; ++nt) {
            const float* __restrict__ wl = Wl + (long long)(nt * 16 + lo16) * 64 + c0;
            const float* __restrict__ wr = Wr + (long long)(nt * 16 + lo16) * 64 + c0;
            v2f bl; bl[0] = wl[0]; bl[1] = wl[1];
            v2f br; br[0] = wr[0]; br[1] = wr[1];
            // D = A*B + C ; args: (neg_a, A, neg_b, B, c_mod, C, reuse_a, reuse_b)
            acc[nt] = __builtin_amdgcn_wmma_f32_16x16x4_f32(
                false, aA, false, bl, (short)0, acc[nt], false, false);
            acc[nt] = __builtin_amdgcn_wmma_f32_16x16x4_f32(
                false, aX, false, br, (short)0, acc[nt], false, false);
        }
    }

    // --- epilogue: + bias, LayerNorm over 64 features, ReLU, store ---------
    float bv[4], lw[4], lb[4];
#pragma unroll
    for (int nt = 0; nt < 4; ++nt) {
        const int n = nt * 16 + lo16;
        bv[nt] = bias[n]; lw[nt] = lnw[n]; lb[nt] = lnb[n];
    }
#pragma unroll
    for (int nt = 0; nt < 4; ++nt)
#pragma unroll
        for (int r = 0; r < 8; ++r) acc[nt][r] += bv[nt];

    const int rowBase = tile * 16;
#pragma unroll
    for (int r = 0; r < 8; ++r) {
        // row (rowBase + r + 8*hi): its 64 values live in acc[0..3][r] across
        // this lane's 16-lane half -> reduce with xor masks 1,2,4,8.
        float s  = acc[0][r] + acc[1][r] + acc[2][r] + acc[3][r];
        float ss = acc[0][r] * acc[0][r] + acc[1][r] * acc[1][r]
                 + acc[2][r] * acc[2][r] + acc[3][r] * acc[3][r];
#pragma unroll
        for (int m = 1; m < 16; m <<= 1) {
            s  += __shfl_xor(s,  m, 32);
            ss += __shfl_xor(ss, m, 32);
        }
        const float mean = s * (1.0f / 64.0f);
        const float var  = ss * (1.0f / 64.0f) - mean * mean;
        const float rstd = rsqrtf(var + LN_EPS);

        const int row = rowBase + r + hi * 8;
        if (row < N) {
            float* __restrict__ orow = out + (long long)row * 64;
#pragma unroll
            for (int nt = 0; nt < 4; ++nt) {
                float v = (acc[nt][r] - mean) * rstd * lw[nt] + lb[nt];
                orow[nt * 16 + lo16] = fmaxf(v, 0.0f);
            }
        }
    }
}

// ---------------------------------------------------------------------------
extern "C" void kernel_launch(void* const* d_in, const int* in_sizes, int n_in,
                              void* d_out, int out_size, void* d_ws, size_t ws_size,
                              hipStream_t stream) {
    const float*     xp  = (const float*)d_in[0];      // [N, 64]
    const float*     Wl  = (const float*)d_in[1];      // [64, 64]
    const float*     Wr  = (const float*)d_in[2];      // [64, 64]
    const float*     b   = (const float*)d_in[3];      // [64]
    const float*     lnw = (const float*)d_in[4];      // [64]
    const float*     lnb = (const float*)d_in[5];      // [64]
    const long long* ei  = (const long long*)d_in[6];  // [2, E] int64

    const int N = in_sizes[0] / 64;
    const int E = in_sizes[6] / 2;

    float* aggsum = (float*)d_ws;                      // [N, 64]
    float* cntp   = aggsum + (long long)N * 64;        // [N]

    // 1) zero scatter workspace (N*65 floats)
    const long long zn = (long long)N * 65;
    zero_ws_kernel<<<(int)((zn + 255) / 256), 256, 0, stream>>>(aggsum, zn);

    // 2) edge gather/scatter: one thread per (edge, feature)
    const long long st = (long long)E * 64;
    edge_scatter_kernel<<<(int)((st + 255) / 256), 256, 0, stream>>>(
        xp, ei, aggsum, cntp, E);

    // 3) fused WMMA GEMM + LayerNorm + ReLU: one wave per 16-node tile
    const int tiles  = (N + 15) / 16;
    const int blocks = (tiles + 7) / 8;                // 8 waves per 256-thread block
    sage_gemm_ln_kernel<<<blocks, 256, 0, stream>>>(
        xp, Wl, Wr, b, lnw, lnb, aggsum, cntp, (float*)d_out, N);
}